// quantumAttentionBlock_62758062129787
// MI455X (gfx1250) — compile-verified
//
#include <hip/hip_runtime.h>
#include <math.h>

// quantumAttentionBlock fused kernel for MI455X (gfx1250, wave32)
// B=512 batches, N=256 tokens, D=16. One block per batch, 512 threads = 16 waves.
// All matmuls on v_wmma_f32_16x16x4_f32 (exact f32, matches reference numerics).

#define BATCH 512
#define NTOK  256
#define DIM   16
#define FMAXF 3.402823466e+38f
#define STG_W 68   // staging row stride (16x64 G tile + pad -> conflict-free K-strided reads)

typedef __attribute__((ext_vector_type(2))) float v2f;
typedef __attribute__((ext_vector_type(8))) float v8f;

static __device__ __forceinline__ v8f wmma4(v2f a, v2f b, v8f c) {
  // 8-arg form per probe: (neg_a, A, neg_b, B, c_mod, C, reuse_a, reuse_b)
  return __builtin_amdgcn_wmma_f32_16x16x4_f32(false, a, false, b, (short)0, c, false, false);
}

static __device__ __forceinline__ float safe_acos(float u) {
  // matches jnp: |u|<=1 -> acos(clip(u)); else (incl. NaN) -> 0
  if (fabsf(u) <= 1.0f) {
    float cu = fminf(fmaxf(u, -1.0f), 1.0f);
    return acosf(cu);
  }
  return 0.0f;
}

static __device__ __forceinline__ float nan_to_num(float v) {
  if (v != v) return 0.0f;
  if (v >  FMAXF) return  FMAXF;
  if (v < -FMAXF) return -FMAXF;
  return v;
}

__global__ __launch_bounds__(512) void qattn_kernel(
    const float* __restrict__ x,     const float* __restrict__ ln1g, const float* __restrict__ ln1b,
    const float* __restrict__ angles,const float* __restrict__ ln2g, const float* __restrict__ ln2b,
    const float* __restrict__ w1,    const float* __restrict__ b1,
    const float* __restrict__ w2,    const float* __restrict__ b2,
    float* __restrict__ out)
{
  extern __shared__ float smem[];
  float* wS   = smem;                  // [16][16]   butterfly W
  float* aS   = smem + 256;            // [256][17]  amplitudes a (padded)
  float* waS  = aS  + 256*17;          // [256][17]  Wa (padded)
  float* vS   = waS + 256*17;          // [256][16]  |Wa|
  float* stgB = vS  + 256*16;          // 16 waves * [16][STG_W] staging

  const int tid = threadIdx.x;
  const int b   = blockIdx.x;

  // ================= Phase 1: per-token state prep + butterfly W =================
  float av[16];
  if (tid < NTOK) {
    const float* xr = x + ((size_t)b * NTOK + tid) * DIM;
    float xi[16];
    float s1 = 0.f, s2 = 0.f;
#pragma unroll
    for (int i = 0; i < 16; i++) { xi[i] = xr[i]; s1 += xi[i]; s2 += xi[i]*xi[i]; }
    float mu  = s1 * (1.f/16.f);
    float var = s2 * (1.f/16.f) - mu*mu;
    float rin = rsqrtf(var + 1e-5f);
    float inp[16];
#pragma unroll
    for (int i = 0; i < 16; i++) inp[i] = ((xi[i]-mu)*rin*ln1g[i] + ln1b[i]) * 0.5f;

    // recursive spherical decomposition (15 serial acos/sin steps)
    float alpha[15];
    alpha[0] = safe_acos(inp[0]);
    float tacc = 1.f, aprev = alpha[0];
#pragma unroll
    for (int i = 1; i < 15; i++) {
      float sv = sinf(aprev);
      float iv = nan_to_num(1.f / sv);
      tacc *= iv;
      aprev = safe_acos(inp[i] * tacc);
      alpha[i] = aprev;
    }
    // amplitudes: a_i = cos(alpha_i) * prod_{k<i} sin(alpha_k)
    float pre = 1.f;
#pragma unroll
    for (int i = 0; i < 15; i++) { av[i] = pre * cosf(alpha[i]); pre *= sinf(alpha[i]); }
    av[15] = pre;
#pragma unroll
    for (int i = 0; i < 16; i++) aS[tid*17 + i] = av[i];
  } else if (tid < NTOK + 16) {
    // 16 threads build the shared butterfly matrix, one column each
    int e = tid - NTOK;
    float col[16];
#pragma unroll
    for (int i = 0; i < 16; i++) col[i] = (i == e) ? 1.f : 0.f;
#pragma unroll
    for (int s = 0; s < 4; s++) {
      const int step = 1 << s;
#pragma unroll
      for (int p = 0; p < 8; p++) {
        int i = (p / step) * (2 * step) + (p % step);
        int j = i + step;
        float ang = angles[s*8 + p];
        float c = cosf(ang), sn = sinf(ang);
        float mi = col[i], mj = col[j];
        col[i] =  c*mi + sn*mj;
        col[j] = -sn*mi + c*mj;
      }
    }
#pragma unroll
    for (int d = 0; d < 16; d++) wS[d*16 + e] = col[d];
  }
  __syncthreads();

  if (tid < NTOK) {
#pragma unroll
    for (int d = 0; d < 16; d++) {
      float s = 0.f;
#pragma unroll
      for (int e = 0; e < 16; e++) s += wS[d*16 + e] * av[e];
      waS[tid*17 + d] = s;
      vS[tid*16 + d]  = fabsf(s);
    }
  }
  __syncthreads();

  // ================= Phase 2: attention, wave wv owns row-tile of 16 tokens =====
  const int lane = tid & 31;
  const int wv   = tid >> 5;      // row tile I = wave id (0..15)
  const int nH   = lane & 15;     // N / M index within half
  const int kh   = lane >> 4;     // half id: K-offset 2*kh in A/B frags, row offset 8*kh in C
  float* stg = stgB + wv * 16 * STG_W;

  // A-fragments of Wa_I (reused across all 16 J tiles)
  float a0[4], a1[4];
#pragma unroll
  for (int ks = 0; ks < 4; ks++) {
    int k = ks*4 + 2*kh;
    a0[ks] = waS[(wv*16 + nH)*17 + k];
    a1[ks] = waS[(wv*16 + nH)*17 + k + 1];
  }

  v8f  oacc = {};                 // unnormalized attn @ vxT accumulator (C layout)
  float rs[8] = {0,0,0,0,0,0,0,0};// per-lane partial row sums of exp

  // |xwx| <= 1 (unit amplitudes, orthogonal W) => softmax needs no max-shift: single pass
#pragma unroll 1
  for (int J = 0; J < 16; J++) {
    v8f sAcc = {};
#pragma unroll
    for (int ks = 0; ks < 4; ks++) {
      int k = ks*4 + 2*kh;
      v2f A  = { a0[ks], a1[ks] };
      v2f Bv = { aS[(J*16 + nH)*17 + k], aS[(J*16 + nH)*17 + k + 1] };
      sAcc = wmma4(A, Bv, sAcc);   // S_IJ = Wa_I . a_J^T
    }
    float pv[8];
#pragma unroll
    for (int r = 0; r < 8; r++) { pv[r] = expf(fabsf(sAcc[r])); rs[r] += pv[r]; }
    // stage P tile (C layout -> row major) for A-fragment reload; same-wave LDS is in-order
#pragma unroll
    for (int r = 0; r < 8; r++) stg[(r + 8*kh)*STG_W + nH] = pv[r];
    __builtin_amdgcn_wave_barrier();
#pragma unroll
    for (int ks = 0; ks < 4; ks++) {
      int k = ks*4 + 2*kh;
      v2f PA = { stg[nH*STG_W + k], stg[nH*STG_W + k + 1] };
      v2f VB = { vS[(J*16 + k)*16 + nH], vS[(J*16 + k + 1)*16 + nH] };
      oacc = wmma4(PA, VB, oacc);  // O += P_IJ . vxT_J
    }
    __builtin_amdgcn_wave_barrier();
  }

  // row-sum reduction across the 16 lanes of each half, normalize, add residual x
  float xn[8];
#pragma unroll
  for (int r = 0; r < 8; r++) {
#pragma unroll
    for (int m = 1; m < 16; m <<= 1) rs[r] += __shfl_xor(rs[r], m, 32);
    oacc[r] *= (1.0f / rs[r]);
    int tok = wv*16 + r + 8*kh;
    float xv = x[((size_t)b * NTOK + tok) * DIM + nH];
    xn[r] = xv + oacc[r];
  }

  // ================= Phase 3: MLP (LN2 -> 16x64 gelu -> 64x16) + residual ========
  const float g2  = ln2g[nH];
  const float bb2 = ln2b[nH];
  const float bo  = b2[nH];

#pragma unroll
  for (int r = 0; r < 8; r++) {
    float v = xn[r];
    float s1 = v, s2 = v*v;
#pragma unroll
    for (int m = 1; m < 16; m <<= 1) { s1 += __shfl_xor(s1, m, 32); s2 += __shfl_xor(s2, m, 32); }
    float mu  = s1 * (1.f/16.f);
    float var = s2 * (1.f/16.f) - mu*mu;
    float h = (v - mu) * rsqrtf(var + 1e-5f) * g2 + bb2;
    stg[(r + 8*kh)*STG_W + nH] = h;
  }
  __builtin_amdgcn_wave_barrier();

  float h0[4], h1[4];
#pragma unroll
  for (int ks = 0; ks < 4; ks++) {
    int k = ks*4 + 2*kh;
    h0[ks] = stg[nH*STG_W + k];
    h1[ks] = stg[nH*STG_W + k + 1];
  }
#pragma unroll
  for (int n2 = 0; n2 < 4; n2++) {
    v8f g = {};
#pragma unroll
    for (int ks = 0; ks < 4; ks++) {
      int k = ks*4 + 2*kh;
      v2f A  = { h0[ks], h1[ks] };
      v2f Bv = { w1[k*64 + n2*16 + nH], w1[(k+1)*64 + n2*16 + nH] };
      g = wmma4(A, Bv, g);         // G = H . w1  (16x64)
    }
    float bb = b1[n2*16 + nH];
#pragma unroll
    for (int r = 0; r < 8; r++) {
      float v = g[r] + bb;
      v = 0.5f * v * (1.0f + erff(v * 0.70710678118654752f));  // exact GELU
      stg[(r + 8*kh)*STG_W + n2*16 + nH] = v;
    }
  }
  __builtin_amdgcn_wave_barrier();

  v8f acc2 = {};
#pragma unroll
  for (int ks = 0; ks < 16; ks++) {
    int k = ks*4 + 2*kh;
    v2f A  = { stg[nH*STG_W + k], stg[nH*STG_W + k + 1] };
    v2f Bv = { w2[k*16 + nH], w2[(k+1)*16 + nH] };
    acc2 = wmma4(A, Bv, acc2);     // out_mlp = gelu(G) . w2  (16x16)
  }

#pragma unroll
  for (int r = 0; r < 8; r++) {
    int tok = wv*16 + r + 8*kh;
    out[((size_t)b * NTOK + tok) * DIM + nH] = xn[r] + acc2[r] + bo;
  }
}

extern "C" void kernel_launch(void* const* d_in, const int* in_sizes, int n_in,
                              void* d_out, int out_size, void* d_ws, size_t ws_size,
                              hipStream_t stream) {
  (void)in_sizes; (void)n_in; (void)out_size; (void)d_ws; (void)ws_size;
  const float* x      = (const float*)d_in[0];
  const float* ln1g   = (const float*)d_in[1];
  const float* ln1b   = (const float*)d_in[2];
  const float* angles = (const float*)d_in[3];
  const float* ln2g   = (const float*)d_in[4];
  const float* ln2b   = (const float*)d_in[5];
  const float* w1     = (const float*)d_in[6];
  const float* b1     = (const float*)d_in[7];
  const float* w2     = (const float*)d_in[8];
  const float* b2     = (const float*)d_in[9];
  float* outp         = (float*)d_out;

  // dynamic LDS: W(256) + a(256*17) + Wa(256*17) + vxT(256*16) + 16 waves * 16*STG_W
  const size_t smemFloats = 256 + 256*17 + 256*17 + 256*16 + 16*16*STG_W;
  const size_t smemBytes  = smemFloats * sizeof(float);   // ~119 KB (< 320 KB/WGP)

  qattn_kernel<<<dim3(BATCH), dim3(512), smemBytes, stream>>>(
      x, ln1g, ln1b, angles, ln2g, ln2b, w1, b1, w2, b2, outp);
}